// CANN_apu_42554535969527
// MI455X (gfx1250) — compile-verified
//
#include <hip/hip_runtime.h>

typedef __attribute__((ext_vector_type(2))) float v2f;
typedef __attribute__((ext_vector_type(8))) float v8f;

#define NN 16384
#define STEPS 14            // prcn - 1, prcn = 15 (static python int in reference)
#define BLOCK 128           // 4 wave32 waves
#define ROWS_PER_WAVE 16
#define ROWS_PER_WG 64
#define GRID (NN / ROWS_PER_WG)   // 256 workgroups

// GEMV step: temp = J @ r ; U = temp + Iext ; buf2 = (0.2*U)^2 ;
// partial[wg] = sum_{rows of wg} 0.005*buf2  (deterministic LDS tree)
__global__ __launch_bounds__(BLOCK)
void cann_gemv_step(const float* __restrict__ J,
                    const float* __restrict__ rin,
                    const float* __restrict__ Iext,
                    float* __restrict__ outU,
                    float* __restrict__ buf2,
                    float* __restrict__ partials)
{
    extern __shared__ float smem[];
    float* lds_r   = smem;            // NN floats (64 KB): staged r vector
    float* lds_y   = smem + NN;       // 64 floats: per-row dot results
    float* lds_red = smem + NN + 64;  // 64 floats: reduction scratch

    // Stage r into LDS, coalesced float4 (128 threads x 128 floats each).
    for (int i = threadIdx.x; i < NN / 4; i += BLOCK)
        ((float4*)lds_r)[i] = ((const float4*)rin)[i];
    __syncthreads();

    const int lane = threadIdx.x & 31;
    const int wave = threadIdx.x >> 5;
    const int half = lane >> 4;      // 0: lanes 0-15 (K0/K1), 1: lanes 16-31 (K2/K3)
    const int mrow = lane & 15;      // matrix row within the 16-row tile
    const int rowBase = blockIdx.x * ROWS_PER_WG + wave * ROWS_PER_WAVE;

    // A-matrix 16x4 f32 layout: lane L holds J[row][k0 + 2*half .. +1] in 2 VGPRs.
    const float* __restrict__ Jrow = J + (size_t)(rowBase + mrow) * NN + 2 * half;

    v8f c0 = {}; v8f c1 = {}; v8f c2 = {}; v8f c3 = {};

#pragma unroll 4
    for (int k0 = 0; k0 < NN; k0 += 16) {
        // WGP-scope speculative prefetch ~8KB ahead in this row, one per
        // 128B cacheline (32 f32) per row; keeps the 8x line reuse in L0.
        if ((k0 & 31) == 0)
            __builtin_prefetch((const void*)(Jrow + k0 + 2048), 0, 3);

        v2f a0 = *(const v2f*)(Jrow + k0 + 0);
        v2f a1 = *(const v2f*)(Jrow + k0 + 4);
        v2f a2 = *(const v2f*)(Jrow + k0 + 8);
        v2f a3 = *(const v2f*)(Jrow + k0 + 12);

        // B 4x16 f32, broadcast r chunk to all 16 columns: half-wave-uniform
        // ds_load -> VGPR0 = r[k0+2*half], VGPR1 = r[k0+2*half+1].
        v2f b0 = *(const v2f*)(lds_r + k0 + 0  + 2 * half);
        v2f b1 = *(const v2f*)(lds_r + k0 + 4  + 2 * half);
        v2f b2 = *(const v2f*)(lds_r + k0 + 8  + 2 * half);
        v2f b3 = *(const v2f*)(lds_r + k0 + 12 + 2 * half);

        c0 = __builtin_amdgcn_wmma_f32_16x16x4_f32(false, a0, false, b0, (short)0, c0, false, false);
        c1 = __builtin_amdgcn_wmma_f32_16x16x4_f32(false, a1, false, b1, (short)0, c1, false, false);
        c2 = __builtin_amdgcn_wmma_f32_16x16x4_f32(false, a2, false, b2, (short)0, c2, false, false);
        c3 = __builtin_amdgcn_wmma_f32_16x16x4_f32(false, a3, false, b3, (short)0, c3, false, false);
    }

    v8f cs = c0 + c1 + c2 + c3;

    // D layout: VGPR v, lane L -> M = v + 8*(L>=16), N = L%16; all N columns equal.
    // Lane 0 holds y[0..7], lane 16 holds y[8..15] of this wave's tile.
    if (mrow == 0) {
        const int base = wave * ROWS_PER_WAVE + half * 8;
#pragma unroll
        for (int v = 0; v < 8; ++v) lds_y[base + v] = cs[v];
    }
    __syncthreads();

    if (threadIdx.x < ROWS_PER_WG) {
        const int row = blockIdx.x * ROWS_PER_WG + threadIdx.x;
        const float temp = lds_y[threadIdx.x];
        const float U = temp + Iext[row];          // alpha=beta=1
        const float t2 = 0.2f * U;
        const float b2 = t2 * t2;
        outU[row]  = U;                            // last step's values survive
        buf2[row]  = b2;
        lds_red[threadIdx.x] = 0.005f * b2;        // k * buf2
    }
    __syncthreads();

    // deterministic fixed-order tree reduction of 64 values
    for (int s = 32; s > 0; s >>= 1) {
        if (threadIdx.x < s) lds_red[threadIdx.x] += lds_red[threadIdx.x + s];
        __syncthreads();
    }
    if (threadIdx.x == 0) partials[blockIdx.x] = lds_red[0];
}

// Reduce 256 partials -> recSum, then r_next = buf2 * (0.04/recSum) / 0.04.
__global__ __launch_bounds__(256)
void cann_normalize(const float* __restrict__ partials,
                    const float* __restrict__ buf2,
                    float* __restrict__ rout,
                    float* __restrict__ outSum,
                    float* __restrict__ outR)
{
    __shared__ float s[256];
    __shared__ float scale_s;
    const int t = threadIdx.x;
    s[t] = partials[t];
    __syncthreads();
    for (int w = 128; w > 0; w >>= 1) {
        if (t < w) s[t] += s[t + w];
        __syncthreads();
    }
    if (t == 0) {
        const float recSum = s[0];
        *outSum = recSum;                          // last step's value survives
        scale_s = (0.04f / recSum) / 0.04f;
    }
    __syncthreads();
    const float scale = scale_s;
    for (int i = t; i < NN; i += 256) {
        const float rn = buf2[i] * scale;
        rout[i] = rn;                              // next step's r
        outR[i] = rn;                              // final net_r region
    }
}

extern "C" void kernel_launch(void* const* d_in, const int* in_sizes, int n_in,
                              void* d_out, int out_size, void* d_ws, size_t ws_size,
                              hipStream_t stream)
{
    (void)in_sizes; (void)n_in; (void)out_size; (void)ws_size;

    const float* Iext = (const float*)d_in[0];   // net_Iext [128,128]
    const float* J    = (const float*)d_in[1];   // J [N,N]
    const float* r0   = (const float*)d_in[2];   // net_r0 (zeros)
    // d_in[3] = prcn (device int scalar); static 15 in reference -> 14 steps.

    float* out    = (float*)d_out;
    float* outU   = out;             // [0, N)     net_Us[-1]
    float* outSum = out + NN;        // [N]        sums[-1]
    float* outR   = out + NN + 1;    // [N+1, 2N+1) final net_r

    float* ws    = (float*)d_ws;
    float* rbuf  = ws;               // N floats
    float* buf2  = ws + NN;          // N floats
    float* parts = ws + 2 * NN;      // GRID floats

    const size_t smem = (size_t)(NN + 128) * sizeof(float);  // ~66 KB < 320 KB WGP LDS

    const float* rin = r0;
    for (int step = 0; step < STEPS; ++step) {
        hipLaunchKernelGGL(cann_gemv_step, dim3(GRID), dim3(BLOCK), smem, stream,
                           J, rin, Iext, outU, buf2, parts);
        hipLaunchKernelGGL(cann_normalize, dim3(1), dim3(256), 0, stream,
                           parts, buf2, rbuf, outSum, outR);
        rin = rbuf;
    }
}